// Lightgcn_62182536511791
// MI455X (gfx1250) — compile-verified
//
#include <hip/hip_runtime.h>
#include <hip/hip_bf16.h>
#include <stdint.h>

#define THREADS 256

#if defined(__AMDGCN__) && \
    __has_builtin(__builtin_amdgcn_global_load_async_to_lds_b64) && \
    __has_builtin(__builtin_amdgcn_s_wait_asynccnt)
#define USE_ASYNC_LDS 1
#else
#define USE_ASYNC_LDS 0
#endif

#if USE_ASYNC_LDS
typedef int lgcn_v2i __attribute__((vector_size(8)));
// b64 async copy: global (AS1) -> LDS (AS3); parameter type per hipcc diagnostic
// is a pointer to a 2-int vector.
#define LGCN_ASYNC_COPY_B64(gsrc, ldst)                                       \
    __builtin_amdgcn_global_load_async_to_lds_b64(                            \
        (__attribute__((address_space(1))) lgcn_v2i*)(gsrc),                  \
        (__attribute__((address_space(3))) lgcn_v2i*)(ldst), 0, 0)
#endif

// ---------------------------------------------------------------------------
// K0: zero the degree/cursor buffer
// ---------------------------------------------------------------------------
__global__ void lgcn_zero_u32(unsigned int* __restrict__ p, int n) {
    int i = blockIdx.x * blockDim.x + threadIdx.x;
    if (i < n) p[i] = 0u;
}

// ---------------------------------------------------------------------------
// K1: in-degree via integer atomics (deterministic)
// ---------------------------------------------------------------------------
__global__ void lgcn_degree(const int* __restrict__ col,
                            unsigned int* __restrict__ deg, int E) {
    int e = blockIdx.x * blockDim.x + threadIdx.x;
    if (e >= E) return;
    __builtin_prefetch(col + e + 4096, 0, 1);
    atomicAdd(&deg[col[e]], 1u);
}

// ---------------------------------------------------------------------------
// K2: dis[i] = deg>0 ? rsqrt(deg) : 0   (must run BEFORE scan clobbers deg)
// ---------------------------------------------------------------------------
__global__ void lgcn_dis(const unsigned int* __restrict__ deg,
                         float* __restrict__ dis, int n) {
    int i = blockIdx.x * blockDim.x + threadIdx.x;
    if (i >= n) return;
    unsigned int d = deg[i];
    dis[i] = d ? __frsqrt_rn((float)d) : 0.0f;
}

// ---------------------------------------------------------------------------
// K3: single-block exclusive scan of deg -> offs[0..n]; also re-initializes
//     deg[i] = offs[i] so it doubles as the scatter cursor array.
// ---------------------------------------------------------------------------
__global__ void lgcn_scan(unsigned int* __restrict__ deg,
                          unsigned int* __restrict__ offs, int n) {
    __shared__ unsigned int tmp[1024];
    __shared__ unsigned int carry_s;
    const int t = threadIdx.x;
    if (t == 0) carry_s = 0u;
    __syncthreads();
    for (int base = 0; base < n; base += 1024) {
        int i = base + t;
        unsigned int v = (i < n) ? deg[i] : 0u;
        tmp[t] = v;
        __syncthreads();
        for (int off = 1; off < 1024; off <<= 1) {
            unsigned int u = (t >= off) ? tmp[t - off] : 0u;
            __syncthreads();
            tmp[t] += u;
            __syncthreads();
        }
        unsigned int incl = tmp[t];
        unsigned int chunk_total = tmp[1023];
        unsigned int carry = carry_s;
        if (i < n) {
            unsigned int o = carry + incl - v;  // exclusive prefix
            offs[i] = o;
            deg[i]  = o;  // cursor init
        }
        __syncthreads();
        if (t == 0) carry_s = carry + chunk_total;
        __syncthreads();
    }
    if (t == 0) offs[n] = carry_s;
}

// ---------------------------------------------------------------------------
// K4: scatter edges into CSR buckets as (src_row, w) 8-byte pairs
// ---------------------------------------------------------------------------
__global__ void lgcn_scatter(const int* __restrict__ row,
                             const int* __restrict__ col,
                             const float* __restrict__ dis,
                             unsigned int* __restrict__ cursor,
                             int2* __restrict__ csr, int E) {
    int e = blockIdx.x * blockDim.x + threadIdx.x;
    if (e >= E) return;
    __builtin_prefetch(row + e + 4096, 0, 1);
    __builtin_prefetch(col + e + 4096, 0, 1);
    int r = row[e], c = col[e];
    float w = dis[r] * dis[c];
    unsigned int pos = atomicAdd(&cursor[c], 1u);
    csr[pos] = make_int2(r, __float_as_int(w));
}

// ---------------------------------------------------------------------------
// K5/K6: propagation. One wave32 per target node; lane owns features
//        [2*lane, 2*lane+1]. Edge list streamed through an LDS double buffer
//        filled with CDNA5 async global->LDS copies; each edge costs the
//        wave one coalesced 256B b64 gather (L2-resident x) + 2 FMAs/lane.
//   layer==0: xout = A x           ; out = a0*xin + a1*xout
//   layer==1: out += a2 * (A xin)  ; (xout unused)
// ---------------------------------------------------------------------------
__global__ void lgcn_prop(const float* __restrict__ xin,
                          const unsigned int* __restrict__ offs,
                          const int2* __restrict__ csr,
                          float* __restrict__ xout,
                          float* __restrict__ out,
                          const float* __restrict__ alpha,
                          int n, int layer) {
    int gt   = blockIdx.x * blockDim.x + threadIdx.x;
    int node = gt >> 5;
    int lane = gt & 31;
    if (node >= n) return;

    const unsigned int s    = offs[node];
    const unsigned int epos = offs[node + 1];
    float accx = 0.0f, accy = 0.0f;

#if USE_ASYNC_LDS
    __shared__ int2 stage[THREADS / 32][2][32];
    const int wv = threadIdx.x >> 5;
    const int nch = (int)((epos - s + 31u) >> 5);

    // prologue: async-issue chunk 0 (lane 0 always in-range for a valid chunk)
    if (nch > 0) {
        unsigned int idx = s + (unsigned int)lane;
        if (idx < epos) {
            LGCN_ASYNC_COPY_B64(csr + idx, &stage[wv][0][lane]);
        }
    }
    for (int k = 0; k < nch; ++k) {
        const int slot = k & 1;
        if (k + 1 < nch) {
            unsigned int idx = s + (unsigned int)(k + 1) * 32u + (unsigned int)lane;
            __builtin_prefetch(csr + idx + 32, 0, 1);
            if (idx < epos) {
                LGCN_ASYNC_COPY_B64(csr + idx, &stage[wv][slot ^ 1][lane]);
            }
            __builtin_amdgcn_s_wait_asynccnt(1);  // chunk k complete (in-order)
        } else {
            __builtin_amdgcn_s_wait_asynccnt(0);
        }
        asm volatile("" ::: "memory");
        unsigned int rem = epos - (s + (unsigned int)k * 32u);
        int valid = rem < 32u ? (int)rem : 32;
        for (int j = 0; j < valid; ++j) {
            int2 p = stage[wv][slot][j];                 // LDS broadcast read
            float w = __int_as_float(p.y);
            const float* xr = xin + (size_t)p.x * 64u;
            float2 v = *(const float2*)(xr + 2 * lane);  // global_load_b64, coalesced
            accx = fmaf(w, v.x, accx);
            accy = fmaf(w, v.y, accy);
        }
    }
#else
    for (unsigned int base = s; base < epos; base += 32u) {
        __builtin_prefetch(csr + base + 64, 0, 1);
        unsigned int idx = base + (unsigned int)lane;
        int2 p = (idx < epos) ? csr[idx] : make_int2(0, 0);
        unsigned int rem = epos - base;
        int valid = rem < 32u ? (int)rem : 32;
        for (int j = 0; j < valid; ++j) {
            int r   = __shfl(p.x, j, 32);
            float w = __int_as_float(__shfl(p.y, j, 32));
            const float* xr = xin + (size_t)r * 64u;
            float2 v = *(const float2*)(xr + 2 * lane);
            accx = fmaf(w, v.x, accx);
            accy = fmaf(w, v.y, accy);
        }
    }
#endif

    const size_t o = (size_t)node * 64u + (size_t)(2 * lane);
    if (layer == 0) {
        float a0 = alpha[0], a1 = alpha[1];
        xout[o]     = accx;
        xout[o + 1] = accy;
        out[o]      = fmaf(a1, accx, a0 * xin[o]);
        out[o + 1]  = fmaf(a1, accy, a0 * xin[o + 1]);
    } else {
        float a2 = alpha[2];
        out[o]     += a2 * accx;
        out[o + 1] += a2 * accy;
    }
}

// ---------------------------------------------------------------------------
extern "C" void kernel_launch(void* const* d_in, const int* in_sizes, int n_in,
                              void* d_out, int out_size, void* d_ws, size_t ws_size,
                              hipStream_t stream) {
    const float* x     = (const float*)d_in[0];
    const int*   ei    = (const int*)d_in[1];
    const float* alpha = (const float*)d_in[2];
    float*       out   = (float*)d_out;

    const int N = in_sizes[0] / 64;
    const int E = in_sizes[1] / 2;
    const int* row = ei;       // edge_index[0] = sources
    const int* col = ei + E;   // edge_index[1] = targets

    // workspace layout (256B-aligned slices)
    char* ws = (char*)d_ws;
    size_t off = 0;
    auto carve = [&](size_t bytes) -> void* {
        void* p = ws + off;
        off = (off + bytes + 255u) & ~(size_t)255u;
        return p;
    };
    unsigned int* deg_cur = (unsigned int*)carve(sizeof(unsigned int) * (size_t)N);
    unsigned int* offs    = (unsigned int*)carve(sizeof(unsigned int) * (size_t)(N + 1));
    float*        dis     = (float*)carve(sizeof(float) * (size_t)N);
    int2*         csr     = (int2*)carve(sizeof(int2) * (size_t)E);
    float*        x1      = (float*)carve(sizeof(float) * (size_t)N * 64u);
    (void)ws_size; (void)n_in; (void)out_size;

    const int gN  = (N + THREADS - 1) / THREADS;
    const int gE  = (E + THREADS - 1) / THREADS;
    const int gP  = (N * 32 + THREADS - 1) / THREADS;  // one wave per node

    lgcn_zero_u32<<<gN, THREADS, 0, stream>>>(deg_cur, N);
    lgcn_degree  <<<gE, THREADS, 0, stream>>>(col, deg_cur, E);
    lgcn_dis     <<<gN, THREADS, 0, stream>>>(deg_cur, dis, N);
    lgcn_scan    <<<1, 1024,    0, stream>>>(deg_cur, offs, N);
    lgcn_scatter <<<gE, THREADS, 0, stream>>>(row, col, dis, deg_cur, csr, E);
    // layer 1: x1 = A x ; out = a0*x + a1*x1
    lgcn_prop    <<<gP, THREADS, 0, stream>>>(x,  offs, csr, x1, out, alpha, N, 0);
    // layer 2: out += a2 * (A x1)
    lgcn_prop    <<<gP, THREADS, 0, stream>>>(x1, offs, csr, x1, out, alpha, N, 1);
}